// OverlapPool2d_37323265802443
// MI455X (gfx1250) — compile-verified
//
#include <hip/hip_runtime.h>
#include <stdint.h>

// Problem constants (from reference): [16,64,224,224] fp32, 3x3 window, stride 2.
#define H   224
#define W   224
#define SH  2
#define SW  2
#define HO  111                    // (224-3)/2 + 1
#define WO  111
#define TY  8                      // output rows per block
#define IN_ROWS (TY*SH + 1)        // 17 input rows staged per block
#define NOUT    (TY*WO)            // 888 outputs per block
#define NTHREADS 256               // 8 wave32s

typedef unsigned int v4u __attribute__((ext_vector_type(4)));
typedef unsigned int v8u __attribute__((ext_vector_type(8)));

__global__ __launch_bounds__(NTHREADS)
void overlap_pool2d_kernel(const float* __restrict__ in, float* __restrict__ out)
{
    __shared__ float tile[IN_ROWS * W];

    const int plane = blockIdx.y;             // b*C + c
    const int oy0   = blockIdx.x * TY;        // first output row of this tile
    const int tid   = threadIdx.x;

    const float* in_plane  = in  + (size_t)plane * (size_t)(H * W);
    float*       out_plane = out + (size_t)plane * (size_t)(HO * WO);

    // ---- Stage IN_ROWS x W fp32 with ONE Tensor Data Mover descriptor ----
    // Issued by wave 0 only (TDM ignores EXEC; one D# per workgroup).
    if (tid == 0) {
        // LDS byte offset: low 32 bits of the flat address of an LDS object
        // are the wave-relative LDS byte address (aperture in the high half).
        const unsigned lds_addr = (unsigned)(uintptr_t)(&tile[0]);
        const int      gy0      = oy0 * SH;
        const uint64_t gaddr    = (uint64_t)(in_plane + (size_t)gy0 * W);
        const unsigned rows_rem = (unsigned)(H - gy0);   // clip bottom tile (OOB rows -> 0, never read)

        // D# group 0 (128b): count=1 | lds_addr | global_addr[56:0] | type=2
        v4u g0 = {0, 0, 0, 0};
        g0[0] = 0x1u;                                    // count=1, user descriptor
        g0[1] = lds_addr;                                // lds_addr[31:0]
        g0[2] = (unsigned)(gaddr & 0xffffffffu);         // global_addr[31:0]
        g0[3] = (unsigned)((gaddr >> 32) & 0x1ffffffu)   // global_addr[56:32]
              | (2u << 30);                              // type = 2 ("image")

        // D# group 1 (256b): wg_mask=0 (not in cluster), data_size=4B,
        // tensor = 224 x rows_rem, tile = 224 x 17, dim0_stride = 224.
        v8u g1 = {0, 0, 0, 0, 0, 0, 0, 0};
        g1[0] = (2u << 16);                              // data_size=2 (4 bytes)
        g1[1] = ((unsigned)W & 0xffffu) << 16;           // tensor_dim0[15:0]=224
        g1[2] = (rows_rem & 0xffffu) << 16;              // tensor_dim0[31:16]=0 | tensor_dim1[15:0]
        g1[3] = ((unsigned)W) << 16;                     // tensor_dim1[31:16]=0 | tile_dim0=224
        g1[4] = (unsigned)IN_ROWS;                       // tile_dim1=17, tile_dim2=0 (2D)
        g1[5] = (unsigned)W;                             // tensor_dim0_stride[31:0]=224
        g1[6] = 0;                                       // stride hi / dim1_stride lo
        g1[7] = 0;                                       // dim1_stride hi

        // 2-operand form: VADDR2/VADDR3 = NULL (tensor up to 2D).
        asm volatile("tensor_load_to_lds %0, %1"
                     :: "s"(g0), "s"(g1)
                     : "memory");
        __builtin_amdgcn_s_wait_tensorcnt(0);            // drain TENSORcnt (issuing wave)
    }
    __syncthreads();                                     // publish LDS to all 8 waves

    // ---- Compute: min/max normalize, product, denormalize ----
#pragma unroll
    for (int i = 0; i < (NOUT + NTHREADS - 1) / NTHREADS; ++i) {
        int idx = tid + i * NTHREADS;
        if (idx < NOUT) {
            int oyl = idx / WO;
            int oy  = oy0 + oyl;
            if (oy < HO) {
                int ox = idx % WO;
                const float* r0 = &tile[(oyl * SH + 0) * W + ox * SW];
                const float* r1 = &tile[(oyl * SH + 1) * W + ox * SW];
                const float* r2 = &tile[(oyl * SH + 2) * W + ox * SW];
                float v0 = r0[0], v1 = r0[1], v2 = r0[2];
                float v3 = r1[0], v4 = r1[1], v5 = r1[2];
                float v6 = r2[0], v7 = r2[1], v8 = r2[2];

                float mn = fminf(fminf(fminf(fminf(v0, v1), fminf(v2, v3)),
                                       fminf(fminf(v4, v5), fminf(v6, v7))), v8);
                float mx = fmaxf(fmaxf(fmaxf(fmaxf(v0, v1), fmaxf(v2, v3)),
                                       fmaxf(fmaxf(v4, v5), fmaxf(v6, v7))), v8);

                float d   = mx - mn;
                float inv = 1.0f / d;           // one rcp instead of 9 divides
                float p = (v0 - mn) * inv;
                p *= (v1 - mn) * inv;
                p *= (v2 - mn) * inv;
                p *= (v3 - mn) * inv;
                p *= (v4 - mn) * inv;
                p *= (v5 - mn) * inv;
                p *= (v6 - mn) * inv;
                p *= (v8 - mn) * inv;
                p *= (v7 - mn) * inv;
                float res = p * d + mn;         // exactly mn in all non-degenerate cases

                // Output is streamed once and never re-read: bypass near caches.
                __builtin_nontemporal_store(res, &out_plane[oy * WO + ox]);
            }
        }
    }
}

extern "C" void kernel_launch(void* const* d_in, const int* in_sizes, int n_in,
                              void* d_out, int out_size, void* d_ws, size_t ws_size,
                              hipStream_t stream) {
    (void)n_in; (void)out_size; (void)d_ws; (void)ws_size;
    const float* in  = (const float*)d_in[0];
    float*       out = (float*)d_out;

    const int planes    = in_sizes[0] / (H * W);        // 16*64 = 1024
    const int row_tiles = (HO + TY - 1) / TY;           // 14

    dim3 grid(row_tiles, planes);
    dim3 block(NTHREADS);
    overlap_pool2d_kernel<<<grid, block, 0, stream>>>(in, out);
}